// MultiHeadSelfAttention_71416716197933
// MI455X (gfx1250) — compile-verified
//
#include <hip/hip_runtime.h>

typedef float v2f __attribute__((ext_vector_type(2)));
typedef float v8f __attribute__((ext_vector_type(8)));

#define WS7 7
#define NH 4
#define HD 32
#define EMB 128
#define EMB3 384
#define IMG 112
#define NWIN_SIDE 16
#define NWIN 256
#define NTOK 49
#define BATCH 16
#define MROWS (BATCH * IMG * IMG)   // 200704

__device__ __forceinline__ v8f wmma4(v2f a, v2f b, v8f c) {
    // D = A(16x4 f32) * B(4x16 f32) + C(16x16 f32)
    return __builtin_amdgcn_wmma_f32_16x16x4_f32(
        /*neg_a=*/false, a, /*neg_b=*/false, b,
        /*c_mod=*/(short)0, c, /*reuse_a=*/false, /*reuse_b=*/false);
}

// Generic pointers to LDS on AMDGPU carry the LDS byte offset in the low
// 32 bits (shared aperture lives in the high bits), so truncation yields the
// LDS address operand the async-DMA instruction wants.
__device__ __forceinline__ unsigned lds_off_u32(const void* p) {
    return (unsigned)(unsigned long long)p;
}

// 16B per lane, global -> LDS, tracked by ASYNCcnt (CDNA5 async path).
__device__ __forceinline__ void async_copy_b128(void* lds_dst, const void* gsrc) {
    unsigned l = lds_off_u32(lds_dst);
    asm volatile("global_load_async_to_lds_b128 %0, %1, off"
                 :: "v"(l), "v"(gsrc)
                 : "memory");
}

__device__ __forceinline__ void wait_async0() {
    asm volatile("s_wait_asynccnt 0x0" ::: "memory");
}

// ---------------------------------------------------------------------------
// Kernel 1: QKV GEMM (M=200704, K=128, N=384) with fused window-partition
// epilogue: writes q/k/v into ws laid out [part][b][head][win][49][32].
// Block = 256 threads = 8 waves, arranged 4 (M) x 2 (N), tile 64x32.
// ---------------------------------------------------------------------------
__global__ __launch_bounds__(256) void qkv_gemm_kernel(
    const float* __restrict__ x, const float* __restrict__ wqkv,
    const float* __restrict__ bqkv, float* __restrict__ ws)
{
    __shared__ __align__(16) float Xs[64][EMB];   // 32 KB
    __shared__ __align__(16) float Wt[EMB][32];   // 16 KB

    const int tid = threadIdx.x;
    const int m0  = blockIdx.x * 64;
    const int n0  = blockIdx.y * 32;

    // Stage X tile (64 rows x 128 k = 2048 x B128), contiguous, async -> LDS.
    {
        const float4* src = (const float4*)(x + (size_t)m0 * EMB);
        #pragma unroll
        for (int i = 0; i < 8; ++i) {
            int idx = tid + i * 256;
            async_copy_b128(&Xs[0][0] + idx * 4, src + idx);
        }
    }
    // Stage W tile (128 k x 32 n = 1024 x B128), async -> LDS.
    #pragma unroll
    for (int it = 0; it < 4; ++it) {
        int i  = tid + it * 256;
        int k  = i >> 3;          // 8 float4 per row of 32
        int c4 = i & 7;
        async_copy_b128(&Wt[k][c4 * 4], wqkv + (size_t)k * EMB3 + n0 + c4 * 4);
    }
    wait_async0();
    __syncthreads();

    const int wave = tid >> 5;
    const int lane = tid & 31;
    const int half = lane >> 4;
    const int lr   = lane & 15;
    const int wm   = wave >> 1;   // 0..3 M-tile
    const int wn   = wave & 1;    // 0..1 N-tile

    v8f acc = {};
    const int arow = wm * 16 + lr;
    const int bcol = wn * 16 + lr;
    #pragma unroll
    for (int kk = 0; kk < EMB / 4; ++kk) {
        const int kb = kk * 4 + 2 * half;
        v2f a = { Xs[arow][kb], Xs[arow][kb + 1] };
        v2f b = { Wt[kb][bcol], Wt[kb + 1][bcol] };
        acc = wmma4(a, b, acc);
    }

    // Epilogue: bias + scatter to windowed layout.
    const int ncol = n0 + wn * 16 + lr;
    const int part = ncol / EMB;            // 0=q 1=k 2=v
    const int head = (ncol % EMB) / HD;
    const int d    = ncol % HD;
    const float bias = bqkv[ncol];
    const size_t part_stride = (size_t)BATCH * NH * NWIN * NTOK * HD;

    #pragma unroll
    for (int g = 0; g < 8; ++g) {
        int m   = m0 + wm * 16 + g + 8 * half;
        int b_  = m / (IMG * IMG);
        int rem = m % (IMG * IMG);
        int h   = rem / IMG, w = rem % IMG;
        int win = (h / WS7) * NWIN_SIDE + (w / WS7);
        int qi  = (h % WS7) * WS7 + (w % WS7);
        size_t off = (size_t)part * part_stride +
            ((((size_t)b_ * NH + head) * NWIN + win) * NTOK + qi) * HD + d;
        ws[off] = acc[g] + bias;
    }
}

// ---------------------------------------------------------------------------
// Kernel 2: window attention. One block per (b, head, window); 128 threads.
// S = q k^T / sqrt(32) + bias; softmax; O = P v. All matmuls via f32 WMMA.
// ---------------------------------------------------------------------------
__global__ __launch_bounds__(128) void win_attn_kernel(
    const float* __restrict__ ws, const float* __restrict__ pos_emb,
    float* __restrict__ out)
{
    __shared__ __align__(16) float qs[64][HD];   // 8 KB, rows 49..63 zero
    __shared__ __align__(16) float ks[64][HD];
    __shared__ __align__(16) float vs[64][HD];
    __shared__ __align__(16) float Ss[64][64];   // 16 KB
    __shared__ float pe[13 * 13];

    const int tid  = threadIdx.x;
    const int win  = blockIdx.x % NWIN;
    const int head = (blockIdx.x / NWIN) % NH;
    const int b_   = blockIdx.x / (NWIN * NH);

    const size_t part_stride = (size_t)BATCH * NH * NWIN * NTOK * HD;
    const size_t base = (((size_t)b_ * NH + head) * NWIN + win) * (NTOK * HD);

    for (int i = tid; i < 13 * 13; i += 128) pe[i] = pos_emb[i];
    // Zero the pad rows (tokens 49..63) with plain LDS stores (disjoint region).
    for (int i = NTOK * HD + tid; i < 64 * HD; i += 128) {
        ((float*)qs)[i] = 0.f;
        ((float*)ks)[i] = 0.f;
        ((float*)vs)[i] = 0.f;
    }
    // Async-stage the contiguous 49x32 q/k/v tiles (392 x B128 each).
    {
        const float* qg = ws + base;
        const float* kg = ws + part_stride + base;
        const float* vg = ws + 2 * part_stride + base;
        for (int i = tid; i < NTOK * HD / 4; i += 128) {
            async_copy_b128(&qs[0][0] + i * 4, qg + i * 4);
            async_copy_b128(&ks[0][0] + i * 4, kg + i * 4);
            async_copy_b128(&vs[0][0] + i * 4, vg + i * 4);
        }
    }
    wait_async0();
    __syncthreads();

    const int wave = tid >> 5;
    const int lane = tid & 31;
    const int half = lane >> 4;
    const int lr   = lane & 15;
    const int arow = wave * 16 + lr;

    // ---- S = q @ k^T (64x64, K=32): wave handles M-tile=wave, 4 N-tiles ----
    v8f acc[4] = {v8f{}, v8f{}, v8f{}, v8f{}};
    #pragma unroll
    for (int kk = 0; kk < HD / 4; ++kk) {
        const int kb = kk * 4 + 2 * half;
        v2f a = { qs[arow][kb], qs[arow][kb + 1] };
        #pragma unroll
        for (int nt = 0; nt < 4; ++nt) {
            const int brow = nt * 16 + lr;          // B[k][n] = ks[n][k]
            v2f b = { ks[brow][kb], ks[brow][kb + 1] };
            acc[nt] = wmma4(a, b, acc[nt]);
        }
    }
    const float scale = 0.17677669529663687f;       // 1/sqrt(32)
    #pragma unroll
    for (int nt = 0; nt < 4; ++nt)
        #pragma unroll
        for (int g = 0; g < 8; ++g)
            Ss[wave * 16 + g + 8 * half][nt * 16 + lr] = acc[nt][g] * scale;
    __syncthreads();

    // ---- bias + softmax, one thread per valid row (0..48), via LDS ----
    if (tid < NTOK) {
        const int r  = tid;
        const int ri = r / WS7, ci = r % WS7;
        float mx = -1e30f;
        for (int j = 0; j < NTOK; ++j) {
            int rj = j / WS7, cj = j % WS7;
            float s = Ss[r][j] + pe[(ri - rj + WS7 - 1) * 13 + (ci - cj + WS7 - 1)];
            Ss[r][j] = s;
            mx = fmaxf(mx, s);
        }
        float sum = 0.f;
        for (int j = 0; j < NTOK; ++j) {
            float e = __expf(Ss[r][j] - mx);
            Ss[r][j] = e;
            sum += e;
        }
        float inv = 1.f / sum;
        for (int j = 0; j < NTOK; ++j) Ss[r][j] *= inv;
        // cols 49..63 multiply zero-padded v rows -> contribute nothing
    }
    __syncthreads();

    // ---- O = P @ v (64x32, K=64): wave handles M-tile=wave, 2 N-tiles ----
    v8f oacc[2] = {v8f{}, v8f{}};
    #pragma unroll
    for (int kk = 0; kk < 64 / 4; ++kk) {
        const int kb = kk * 4 + 2 * half;
        v2f a = { Ss[arow][kb], Ss[arow][kb + 1] };
        #pragma unroll
        for (int nt = 0; nt < 2; ++nt) {
            v2f b = { vs[kb][nt * 16 + lr], vs[kb + 1][nt * 16 + lr] };
            oacc[nt] = wmma4(a, b, oacc[nt]);
        }
    }

    // ---- scatter back to (B, H, W, C) ----
    const int wr = (win / NWIN_SIDE) * WS7;
    const int wc = (win % NWIN_SIDE) * WS7;
    #pragma unroll
    for (int nt = 0; nt < 2; ++nt) {
        const int c = head * HD + nt * 16 + lr;
        #pragma unroll
        for (int g = 0; g < 8; ++g) {
            int r = wave * 16 + g + 8 * half;
            if (r < NTOK) {
                int h = wr + r / WS7;
                int w = wc + r % WS7;
                out[(((size_t)b_ * IMG + h) * IMG + w) * EMB + c] = oacc[nt][g];
            }
        }
    }
}

extern "C" void kernel_launch(void* const* d_in, const int* in_sizes, int n_in,
                              void* d_out, int out_size, void* d_ws, size_t ws_size,
                              hipStream_t stream) {
    const float* x    = (const float*)d_in[0];
    const float* wqkv = (const float*)d_in[1];
    const float* bqkv = (const float*)d_in[2];
    const float* pos  = (const float*)d_in[3];
    float* out = (float*)d_out;
    float* ws  = (float*)d_ws;   // needs 3*16*4*256*49*32 floats (~308 MB)

    dim3 g1(MROWS / 64, EMB3 / 32);
    qkv_gemm_kernel<<<g1, 256, 0, stream>>>(x, wqkv, bqkv, ws);
    win_attn_kernel<<<BATCH * NH * NWIN, 128, 0, stream>>>(ws, pos, out);
}